// PhotoL1_19310172963011
// MI455X (gfx1250) — compile-verified
//
#include <hip/hip_runtime.h>
#include <hip/hip_bf16.h>

// PhotoL1 bilinear-warp loss for MI455X (gfx1250, wave32).
// Bandwidth-bound gather kernel (~120 MB traffic -> ~5us floor @ 23.3 TB/s).
// Pass 1: one block per (batch,row); 256 thr x 4 px = 1024 = W, b128 row loads,
//         12 cached b32 gathers/pixel, wave32 shuffle + LDS reduce -> partial/row.
// Pass 2: one wave per batch; V_WMMA_F32_16X16X4_F32 with A=ones reduces 64
//         partials/instruction (f32 accumulate), xor-shuffle finishes the mean.

typedef __attribute__((ext_vector_type(2))) float v2f;
typedef __attribute__((ext_vector_type(8))) float v8f;

#define B_  8
#define H_  448
#define W_  1024
#define HW_ (H_ * W_)

__global__ __launch_bounds__(256) void photo_l1_rows(
    const float* __restrict__ flow,   // [B,2,H,W]
    const float* __restrict__ imgs,   // [B,6,H,W] (prev 0..2, next 3..5)
    float* __restrict__ partial)      // [B,H] per-row sums
{
    const int y   = blockIdx.x;    // row 0..447
    const int b   = blockIdx.y;    // batch 0..7
    const int tid = threadIdx.x;   // 0..255
    const int x4  = tid * 4;

    const float* f0    = flow + (size_t)b * 2 * HW_ + (size_t)y * W_;
    const float* f1    = f0 + HW_;
    const float* base  = imgs + (size_t)b * 6 * HW_;
    const float* prev0 = base;
    const float* prev1 = base + 1 * HW_;
    const float* prev2 = base + 2 * HW_;
    const float* next0 = base + 3 * HW_;
    const float* next1 = base + 4 * HW_;
    const float* next2 = base + 5 * HW_;

    // Coalesced 128-bit row loads.
    const float4 fu = *(const float4*)(f0 + x4);
    const float4 fv = *(const float4*)(f1 + x4);
    const float4 p0 = *(const float4*)(prev0 + (size_t)y * W_ + x4);
    const float4 p1 = *(const float4*)(prev1 + (size_t)y * W_ + x4);
    const float4 p2 = *(const float4*)(prev2 + (size_t)y * W_ + x4);

    const float* fup = (const float*)&fu;
    const float* fvp = (const float*)&fv;
    const float* p0p = (const float*)&p0;
    const float* p1p = (const float*)&p1;
    const float* p2p = (const float*)&p2;

    float acc = 0.0f;
#pragma unroll
    for (int j = 0; j < 4; ++j) {
        const float px = (float)(x4 + j) + fup[j];
        const float py = (float)y + fvp[j];

        const float x0f = floorf(px);
        const float y0f = floorf(py);
        const float wx1 = px - x0f;
        const float wy1 = py - y0f;
        const float wx0 = 1.0f - wx1;
        const float wy0 = 1.0f - wy1;

        const int xi0 = (int)x0f, yi0 = (int)y0f;
        const int xi1 = xi0 + 1,  yi1 = yi0 + 1;

        const float vx0 = (xi0 >= 0 && xi0 < W_) ? 1.0f : 0.0f;
        const float vx1 = (xi1 >= 0 && xi1 < W_) ? 1.0f : 0.0f;
        const float vy0 = (yi0 >= 0 && yi0 < H_) ? 1.0f : 0.0f;
        const float vy1 = (yi1 >= 0 && yi1 < H_) ? 1.0f : 0.0f;

        const float m00 = wx0 * wy0 * (vx0 * vy0);
        const float m01 = wx1 * wy0 * (vx1 * vy0);
        const float m10 = wx0 * wy1 * (vx0 * vy1);
        const float m11 = wx1 * wy1 * (vx1 * vy1);

        const int xc0 = min(max(xi0, 0), W_ - 1);
        const int xc1 = min(max(xi1, 0), W_ - 1);
        const int yc0 = min(max(yi0, 0), H_ - 1);
        const int yc1 = min(max(yi1, 0), H_ - 1);

        const int i00 = yc0 * W_ + xc0;
        const int i01 = yc0 * W_ + xc1;
        const int i10 = yc1 * W_ + xc0;
        const int i11 = yc1 * W_ + xc1;

        const float msum = m00 + m01 + m10 + m11;
        const float mk   = (msum >= 0.9999f) ? 1.0f : 0.0f;

        const float s0 = next0[i00] * m00 + next0[i01] * m01 +
                         next0[i10] * m10 + next0[i11] * m11;
        const float s1 = next1[i00] * m00 + next1[i01] * m01 +
                         next1[i10] * m10 + next1[i11] * m11;
        const float s2 = next2[i00] * m00 + next2[i01] * m01 +
                         next2[i10] * m10 + next2[i11] * m11;

        acc += fabsf(s0 * mk - p0p[j]);
        acc += fabsf(s1 * mk - p1p[j]);
        acc += fabsf(s2 * mk - p2p[j]);
    }

    // Wave32 cross-lane reduction, then LDS across the 8 waves (deterministic).
#pragma unroll
    for (int off = 16; off > 0; off >>= 1)
        acc += __shfl_xor(acc, off, 32);

    __shared__ float wsum[8];
    const int wid = tid >> 5;
    if ((tid & 31) == 0) wsum[wid] = acc;
    __syncthreads();
    if (tid == 0) {
        float t = 0.0f;
#pragma unroll
        for (int w = 0; w < 8; ++w) t += wsum[w];
        partial[b * H_ + y] = t;
    }
}

// One wave per batch. A = ones(16x4) so D = A*B + C column-sums B:
// each V_WMMA_F32_16X16X4_F32 folds 64 f32 partials into the 16x16 f32
// accumulator; 448 partials = 7 accumulating WMMAs. C/D layout: VGPR0
// lanes 0..15 hold D[0, n=lane] (all rows identical), so a 16-lane
// xor-shuffle of c[0] yields the total.
__global__ __launch_bounds__(32) void photo_l1_finalize(
    const float* __restrict__ partial,  // [B,H]
    float* __restrict__ out)            // [B]
{
    const int b    = blockIdx.x;
    const int lane = threadIdx.x;  // 0..31, EXEC all ones as WMMA requires
    const float* p = partial + b * H_;

    v2f ones;
    ones.x = 1.0f;
    ones.y = 1.0f;
    v8f c = {};

#pragma unroll
    for (int chunk = 0; chunk < H_; chunk += 64) {
        v2f bm;
        bm.x = p[chunk + lane];
        bm.y = p[chunk + 32 + lane];
        // D(16x16,f32) = ones(16x4) * B(4x16) + C  -> emits v_wmma_f32_16x16x4_f32
        c = __builtin_amdgcn_wmma_f32_16x16x4_f32(
                /*neg_a=*/false, ones, /*neg_b=*/false, bm,
                /*c_mod=*/(short)0, c, /*reuse_a=*/false, /*reuse_b=*/false);
    }

    float t = c[0];
#pragma unroll
    for (int off = 1; off < 16; off <<= 1)
        t += __shfl_xor(t, off, 32);

    if (lane == 0) out[b] = t * (1.0f / (float)HW_);
}

extern "C" void kernel_launch(void* const* d_in, const int* in_sizes, int n_in,
                              void* d_out, int out_size, void* d_ws, size_t ws_size,
                              hipStream_t stream) {
    const float* flow = (const float*)d_in[0];  // 'outputs': [B,2,H,W] f32
    const float* imgs = (const float*)d_in[1];  // 'inputs' : [B,6,H,W] f32
    float* out     = (float*)d_out;             // [B] f32
    float* partial = (float*)d_ws;              // B*H floats = 14336 B scratch

    dim3 grid(H_, B_);
    photo_l1_rows<<<grid, 256, 0, stream>>>(flow, imgs, partial);
    photo_l1_finalize<<<B_, 32, 0, stream>>>(partial, out);
}